// SNN_2714419331192
// MI455X (gfx1250) — compile-verified
//
#include <hip/hip_runtime.h>
#include <math.h>

#define B_  16
#define S_  64
#define BH_ 768
#define H_  512
#define N_  128
#define T_  (S_ * H_)      // 32768 sequential LIF steps per batch
#define CHUNK 1024
#define NCHUNK (T_ / CHUNK)  // 32

typedef float v2f __attribute__((ext_vector_type(2)));
typedef float v4f __attribute__((ext_vector_type(4)));
typedef float v8f __attribute__((ext_vector_type(8)));

// ---------------------------------------------------------------------------
// GEMM via V_WMMA_F32_16X16X4_F32: out[M,Nout] = A[M,K] * W[Nout,K]^T + bias
// One wave (32 threads) per 16x16 output tile. EXEC all-ones inside the wave
// (required by WMMA). fp32 matrix path: GEMM cost is negligible vs the spike
// store, and LIF threshold crossings amplify precision error, so no bf16.
// ---------------------------------------------------------------------------
__global__ __launch_bounds__(32) void gemm_wmma_f32(
    const float* __restrict__ A, const float* __restrict__ W,
    const float* __restrict__ bias, float* __restrict__ out,
    int M, int K, int Nout)
{
  const int lane   = threadIdx.x;              // 0..31
  const int tilesN = Nout >> 4;
  const int tileM  = blockIdx.x / tilesN;
  const int tileN  = blockIdx.x % tilesN;
  const int lo     = lane & 15;
  const int hi     = lane >> 4;                // 0: K=0,1  1: K=2,3  (ISA A-layout)

  const int rowA = (tileM << 4) + lo;
  const int colB = (tileN << 4) + lo;
  const float* __restrict__ Arow = A + (size_t)rowA * K + (hi << 1);
  const float* __restrict__ Wrow = W + (size_t)colB * K + (hi << 1);

  v8f c = {};
  for (int kk = 0; kk < K; kk += 4) {
    if ((kk & 63) == 0) {                      // global_prefetch_b8, 256 floats ahead
      __builtin_prefetch(Arow + kk + 256, 0, 3);
      __builtin_prefetch(Wrow + kk + 256, 0, 3);
    }
    v2f a, b;
    a.x = Arow[kk];       // K = kk + 2*hi
    a.y = Arow[kk + 1];   // K = kk + 2*hi + 1
    b.x = Wrow[kk];
    b.y = Wrow[kk + 1];
    // 8 args: (neg_a, A, neg_b, B, c_mod, C, reuse_a, reuse_b)
    c = __builtin_amdgcn_wmma_f32_16x16x4_f32(
        false, a, false, b, (short)0, c, false, false);
  }

  // C/D layout: VGPR i -> M = i (lanes 0-15) / M = i+8 (lanes 16-31), N = lane&15
  const float bv = bias[colB];
#pragma unroll
  for (int i = 0; i < 8; ++i) {
    int r = (tileM << 4) + (hi << 3) + i;
    out[(size_t)r * Nout + colB] = c[i] + bv;
  }
}

// ---------------------------------------------------------------------------
// Inclusive prefix sum of x[b][0..T) -> P[b][0..T). One block of 256 per batch:
// per-thread segment sums, Hillis-Steele block scan in LDS, then re-walk.
// ---------------------------------------------------------------------------
__global__ __launch_bounds__(256) void scan_kernel(
    const float* __restrict__ x, float* __restrict__ P)
{
  __shared__ float lds[256];
  const int b   = blockIdx.x;
  const int tid = threadIdx.x;
  const int SEG = T_ / 256;                    // 128
  const float* __restrict__ xb = x + (size_t)b * T_;
  float* __restrict__ Pb = P + (size_t)b * T_;

  const int base = tid * SEG;
  float s = 0.f;
  for (int j = 0; j < SEG; ++j) s += xb[base + j];
  lds[tid] = s;
  __syncthreads();
  for (int off = 1; off < 256; off <<= 1) {
    float t = (tid >= off) ? lds[tid - off] : 0.f;
    __syncthreads();
    lds[tid] += t;
    __syncthreads();
  }
  float acc = (tid == 0) ? 0.f : lds[tid - 1]; // exclusive base for my segment
  for (int j = 0; j < SEG; ++j) {
    acc += xb[base + j];
    Pb[base + j] = acc;
  }
}

// ---------------------------------------------------------------------------
// Async global->LDS copy of one 1024-float P chunk (ASYNCcnt-tracked DMA path,
// GLOBAL_LOAD_ASYNC_TO_LDS_B128 in GVS mode: saddr base + 32-bit vgpr offset).
// Each of 128 threads moves 2x16B. LDS dest VGPR holds the workgroup-relative
// LDS byte offset (low 32 bits of the generic pointer).
// ---------------------------------------------------------------------------
__device__ __forceinline__ void async_load_chunk(
    const float* __restrict__ Pb, int tN, int n, float* ldsDst)
{
  unsigned ldsa0 = (unsigned)(unsigned long long)(ldsDst + n * 4);
  unsigned ldsa1 = (unsigned)(unsigned long long)(ldsDst + 512 + n * 4);
  unsigned off0  = (unsigned)((tN + n * 4) * sizeof(float));
  unsigned off1  = (unsigned)((tN + 512 + n * 4) * sizeof(float));
  asm volatile("global_load_async_to_lds_b128 %0, %1, %2\n\t"
               "global_load_async_to_lds_b128 %3, %4, %2"
               :: "v"(ldsa0), "v"(off0), "s"(Pb), "v"(ldsa1), "v"(off1)
               : "memory");
}

__device__ __forceinline__ void wait_async_all() {
  asm volatile("s_wait_asynccnt 0x0" ::: "memory");
}

// ---------------------------------------------------------------------------
// LIF via prefix sums: acc_n(t) = P[t] - P[last_reset_n]. One block per batch,
// one thread per neuron (128). P chunks double-buffered in LDS via async DMA;
// per-step osum reduced with ds_add_f32 atomics (spikes only); spike floats
// packed 4-wide (consecutive h) into non-temporal v4f stores (write-once
// 805MB stream must not evict the 192MB L2 working set).
// ---------------------------------------------------------------------------
__global__ __launch_bounds__(128) void lif_kernel(
    const float* __restrict__ P, const float* __restrict__ thresh,
    float* __restrict__ spikes, float* __restrict__ xint)
{
  __shared__ float pch[2][CHUNK];
  __shared__ float osum[2][CHUNK];
  const int b = blockIdx.x;
  const int n = threadIdx.x;
  const float th = thresh[n];
  const float* __restrict__ Pb = P + (size_t)b * T_;
  float base = 0.f;

  // Prologue: DMA chunk 0 into buffer 0; zero both osum buffers.
  async_load_chunk(Pb, 0, n, &pch[0][0]);
  for (int j = n; j < CHUNK; j += N_) { osum[0][j] = 0.f; osum[1][j] = 0.f; }
  wait_async_all();
  __syncthreads();

  for (int c = 0; c < NCHUNK; ++c) {
    const int cur = c & 1;
    const int nb  = cur ^ 1;
    const int t0  = c * CHUNK;
    const int tN  = t0 + CHUNK;

    // Kick off DMA of the next chunk into the other buffer while we compute.
    if (tN < T_) async_load_chunk(Pb, tN, n, &pch[nb][0]);

    for (int j = 0; j < CHUNK; j += 4) {
      const int t = t0 + j;                    // t0,j mult of 4; h stays in-row
      const int s_ = t >> 9;                   // t / H
      const int h_ = t & (H_ - 1);
      v4f v;
#pragma unroll
      for (int u = 0; u < 4; ++u) {
        float p = pch[cur][j + u];
        float d = p - base;
        bool spk = d > th;
        v[u] = spk ? 1.0f : 0.0f;
        if (spk) {
          atomicAdd(&osum[cur][j + u], d);     // ds_add_f32, spikes only
          base = p;                            // reset: acc -> 0
        }
      }
      size_t sidx = ((((size_t)b * S_ + s_) * N_ + n) * H_) + h_;  // 16B aligned
      __builtin_nontemporal_store(v, (v4f*)(spikes + sidx));       // th:NT
    }
    __syncthreads();                           // all osum[cur] atomics complete

    // Write out integrated values; re-zero this buffer for chunk c+2.
    float* __restrict__ Xb = xint + (size_t)b * T_ + t0;
    for (int j = n; j < CHUNK; j += N_) {
      Xb[j] = osum[cur][j];
      osum[cur][j] = 0.f;
    }
    wait_async_all();                          // pch[nb] DMA landed in LDS
    __syncthreads();                           // all threads see pch[nb] + zeros
  }
}

// ---------------------------------------------------------------------------
// pooled[b,s] = tanh(sum_h xint[b,s,h] * wp[h] + bp)
// ---------------------------------------------------------------------------
__global__ __launch_bounds__(256) void pool_kernel(
    const float* __restrict__ xint, const float* __restrict__ wp,
    const float* __restrict__ bp, float* __restrict__ out)
{
  int idx = blockIdx.x * blockDim.x + threadIdx.x;   // 0..B*S-1
  if (idx >= B_ * S_) return;
  const float* __restrict__ row = xint + (size_t)idx * H_;
  float s = 0.f;
  for (int h = 0; h < H_; ++h) s += row[h] * wp[h];
  out[idx] = tanhf(s + bp[0]);
}

// ---------------------------------------------------------------------------
extern "C" void kernel_launch(void* const* d_in, const int* in_sizes, int n_in,
                              void* d_out, int out_size, void* d_ws, size_t ws_size,
                              hipStream_t stream) {
  const float* inputs = (const float*)d_in[0];   // [B,S,BH]
  const float* w1     = (const float*)d_in[1];   // [H,BH]
  const float* b1     = (const float*)d_in[2];   // [H]
  const float* w2     = (const float*)d_in[3];   // [H,H]
  const float* b2     = (const float*)d_in[4];
  const float* w3     = (const float*)d_in[5];
  const float* b3     = (const float*)d_in[6];
  const float* t1     = (const float*)d_in[7];   // [N]
  const float* t2     = (const float*)d_in[8];
  const float* t3     = (const float*)d_in[9];
  const float* wp     = (const float*)d_in[10];  // [1,H]
  const float* bp     = (const float*)d_in[11];  // [1]

  float* out = (float*)d_out;
  float* pooled = out;                                         // [B,S]
  const size_t spikes_per_layer = (size_t)B_ * S_ * N_ * H_;   // 67,108,864
  float* spikes0 = out + (size_t)B_ * S_;
  float* spikes1 = spikes0 + spikes_per_layer;
  float* spikes2 = spikes1 + spikes_per_layer;

  char* ws = (char*)d_ws;
  const size_t actBytes = (size_t)B_ * T_ * sizeof(float);     // 2 MB each
  float* xa   = (float*)(ws);                                  // GEMM output
  float* xint = (float*)(ws + actBytes);                       // LIF integrated
  float* Pfx  = (float*)(ws + 2 * actBytes);                   // prefix sums

  const int M = B_ * S_;                                       // 1024
  const int tiles1 = (M / 16) * (H_ / 16);                     // 2048 waves

  // Layer 1
  gemm_wmma_f32<<<tiles1, 32, 0, stream>>>(inputs, w1, b1, xa, M, BH_, H_);
  scan_kernel<<<B_, 256, 0, stream>>>(xa, Pfx);
  lif_kernel<<<B_, N_, 0, stream>>>(Pfx, t1, spikes0, xint);
  // Layer 2
  gemm_wmma_f32<<<tiles1, 32, 0, stream>>>(xint, w2, b2, xa, M, H_, H_);
  scan_kernel<<<B_, 256, 0, stream>>>(xa, Pfx);
  lif_kernel<<<B_, N_, 0, stream>>>(Pfx, t2, spikes1, xint);
  // Layer 3
  gemm_wmma_f32<<<tiles1, 32, 0, stream>>>(xint, w3, b3, xa, M, H_, H_);
  scan_kernel<<<B_, 256, 0, stream>>>(xa, Pfx);
  lif_kernel<<<B_, N_, 0, stream>>>(Pfx, t3, spikes2, xint);
  // Pool
  pool_kernel<<<(B_ * S_ + 255) / 256, 256, 0, stream>>>(xint, wp, bp, pooled);
}